// LightGCN_55637006353092
// MI455X (gfx1250) — compile-verified
//
#include <hip/hip_runtime.h>
#include <stdint.h>

#define N_USERS  100000
#define N_ITEMS  50000
#define N_NODES  150000
#define EMBED    64
#define NNZ      4000000
#define BATCH    4096

typedef __attribute__((ext_vector_type(2))) float v2f;
typedef __attribute__((ext_vector_type(8))) float v8f;

// ---------------------------------------------------------------------------
// CDNA5 async global->LDS copy (ASYNCcnt path, gfx1250-only instruction).
// lds_byte_off: LDS byte address (low 32 bits of a generic shared pointer).
// ---------------------------------------------------------------------------
__device__ __forceinline__ void async_load_dword_to_lds(uint32_t lds_byte_off,
                                                        uint64_t gaddr) {
    asm volatile("global_load_async_to_lds_b32 %0, %1, off"
                 :: "v"(lds_byte_off), "v"(gaddr)
                 : "memory");
}

__device__ __forceinline__ void wait_asynccnt0() {
    asm volatile("s_wait_asynccnt 0" ::: "memory");
}

// ---------------------------------------------------------------------------
// init: buf0 = concat(embed_user, embed_item); buf1 = 0   (float4 vectorized)
// ---------------------------------------------------------------------------
__global__ void lgcn_init_kernel(const float* __restrict__ eu,
                                 const float* __restrict__ ei,
                                 float* __restrict__ buf0,
                                 float* __restrict__ buf1) {
    const size_t i = (size_t)blockIdx.x * blockDim.x + threadIdx.x; // float4 idx
    const size_t NU4  = (size_t)N_USERS * (EMBED / 4);
    const size_t TOT4 = (size_t)N_NODES * (EMBED / 4);
    if (i >= TOT4) return;
    float4 v = (i < NU4) ? ((const float4*)eu)[i] : ((const float4*)ei)[i - NU4];
    ((float4*)buf0)[i] = v;
    ((float4*)buf1)[i] = make_float4(0.f, 0.f, 0.f, 0.f);
}

__global__ void lgcn_zero_kernel(float4* __restrict__ p) {
    const size_t i = (size_t)blockIdx.x * blockDim.x + threadIdx.x;
    if (i < (size_t)N_NODES * (EMBED / 4))
        p[i] = make_float4(0.f, 0.f, 0.f, 0.f);
}

// ---------------------------------------------------------------------------
// COO SpMM scatter: Eout[row] += a * Ein[col]  (segment_sum via f32 atomics).
// 16 nnz per 256-thread block; 16 lanes x float4 per nnz (coalesced 256B/row).
// Triplets staged into LDS with CDNA5 async load-to-LDS.
// ---------------------------------------------------------------------------
__global__ void lgcn_scatter_kernel(const int*   __restrict__ row_idx,
                                    const int*   __restrict__ col_idx,
                                    const float* __restrict__ a_vals,
                                    const float* __restrict__ Ein,
                                    float*       __restrict__ Eout) {
    __shared__ int   s_row[16];
    __shared__ int   s_col[16];
    __shared__ float s_val[16];

    const int t = threadIdx.x;
    const size_t base = (size_t)blockIdx.x * 16;

    if (t < 48) {
        uint64_t g;
        uint32_t lds;
        if (t < 16) {
            g   = (uint64_t)(uintptr_t)(row_idx + base + t);
            lds = (uint32_t)(uintptr_t)(&s_row[t]);
        } else if (t < 32) {
            g   = (uint64_t)(uintptr_t)(col_idx + base + (t - 16));
            lds = (uint32_t)(uintptr_t)(&s_col[t - 16]);
        } else {
            g   = (uint64_t)(uintptr_t)(a_vals + base + (t - 32));
            lds = (uint32_t)(uintptr_t)(&s_val[t - 32]);
        }
        async_load_dword_to_lds(lds, g);
    }
    wait_asynccnt0();
    __syncthreads();

    const int j = t >> 4;          // nnz within block: 0..15
    const int d = (t & 15) << 2;   // embed dim chunk: 0,4,...,60

    const int   r = s_row[j];
    const int   c = s_col[j];
    const float a = s_val[j];

    const float4 e = *(const float4*)(Ein + (size_t)c * EMBED + d); // L2-resident gather
    float* dst = Eout + (size_t)r * EMBED + d;
    atomicAdd(dst + 0, a * e.x);
    atomicAdd(dst + 1, a * e.y);
    atomicAdd(dst + 2, a * e.z);
    atomicAdd(dst + 3, a * e.w);
}

// ---------------------------------------------------------------------------
// Gather + scale-accumulate epilogue using V_WMMA_F32_16X16X4_F32:
//   out_tile (16x16) (+)= 0.25 * E[srcrow(i)][c0+j]
// Implemented as D = sum_p (0.25*Sel_p) x B_p + C, where Sel_p is the 16x4
// selector picking rows 4p..4p+3.  One 16-row slab per block, 4 waves cover
// the 64 embed columns as 16-col tiles.
//   A layout (16x4 f32): lanes 0-15 -> K=0,1 ; lanes 16-31 -> K=2,3
//   B layout (4x16 f32): vgpr0: K=2*half row at col=lane&15 ; vgpr1: K=2*half+1
//   C/D layout: vgpr v:  lanes 0-15 -> M=v ; lanes 16-31 -> M=v+8
// ---------------------------------------------------------------------------
__global__ void lgcn_gather_wmma_kernel(const float* __restrict__ E,
                                        const int* __restrict__ bu,
                                        const int* __restrict__ bp,
                                        const int* __restrict__ bn,
                                        float* __restrict__ out,
                                        int init) {
    __shared__ int srow[16];

    const int t   = threadIdx.x;
    const int gr0 = blockIdx.x * 16;          // base output row (0..12287)

    if (t < 16) {
        const int g   = gr0 + t;
        const int sec = g >> 12;              // /4096: 0=user,1=pos,2=neg
        const int b   = g & 4095;
        int idx;
        if (sec == 0)      idx = bu[b];
        else if (sec == 1) idx = N_USERS + bp[b];
        else               idx = N_USERS + bn[b];
        srow[t] = idx;
    }
    __syncthreads();

    const int wave = t >> 5;                  // 0..3 -> 16-col tile
    const int lane = t & 31;
    const int half = lane >> 4;
    const int col  = lane & 15;
    const int c0   = wave * 16;
    const int m_a  = lane & 15;               // A-matrix row held by this lane
    const int k0   = 2 * half;                // A/B K base for this half-wave

    v8f c = {};
    if (!init) {
#pragma unroll
        for (int v = 0; v < 8; ++v) {
            const int m = v + 8 * half;
            c[v] = out[(size_t)(gr0 + m) * EMBED + c0 + col];
        }
    }

#pragma unroll
    for (int p = 0; p < 4; ++p) {
        const int r0 = 4 * p + k0;            // tile rows supplied by this lane
        v2f b;
        b.x = E[(size_t)srow[r0]     * EMBED + c0 + col];
        b.y = E[(size_t)srow[r0 + 1] * EMBED + c0 + col];
        v2f a;
        a.x = (m_a == 4 * p + k0)     ? 0.25f : 0.0f;
        a.y = (m_a == 4 * p + k0 + 1) ? 0.25f : 0.0f;
        c = __builtin_amdgcn_wmma_f32_16x16x4_f32(
                /*neg_a=*/false, a, /*neg_b=*/false, b,
                /*c_mod=*/(short)0, c, /*reuse_a=*/false, /*reuse_b=*/false);
    }

#pragma unroll
    for (int v = 0; v < 8; ++v) {
        const int m = v + 8 * half;
        out[(size_t)(gr0 + m) * EMBED + c0 + col] = c[v];
    }
}

// ---------------------------------------------------------------------------
extern "C" void kernel_launch(void* const* d_in, const int* in_sizes, int n_in,
                              void* d_out, int out_size, void* d_ws, size_t ws_size,
                              hipStream_t stream) {
    const int*   batch_user = (const int*)d_in[0];
    const int*   batch_pos  = (const int*)d_in[1];
    const int*   batch_neg  = (const int*)d_in[2];
    const float* embed_user = (const float*)d_in[3];
    const float* embed_item = (const float*)d_in[4];
    const int*   row_idx    = (const int*)d_in[5];
    const int*   col_idx    = (const int*)d_in[6];
    const float* a_vals     = (const float*)d_in[7];
    float*       out        = (float*)d_out;

    float* buf0 = (float*)d_ws;                         // 38.4 MB
    float* buf1 = buf0 + (size_t)N_NODES * EMBED;       // 38.4 MB

    const int vecN    = (N_NODES * EMBED) / 4;          // 2,400,000 float4
    const int vecBlk  = (vecN + 255) / 256;             // 9375
    const int scatBlk = NNZ / 16;                       // 250,000
    const int gathBlk = (3 * BATCH) / 16;               // 768

    // E0 = concat(users, items); other ping buffer zeroed
    lgcn_init_kernel<<<vecBlk, 256, 0, stream>>>(embed_user, embed_item, buf0, buf1);
    // out = 0.25 * E0[batch rows]
    lgcn_gather_wmma_kernel<<<gathBlk, 128, 0, stream>>>(buf0, batch_user, batch_pos,
                                                         batch_neg, out, 1);
    // layer 1: buf0 -> buf1
    lgcn_scatter_kernel<<<scatBlk, 256, 0, stream>>>(row_idx, col_idx, a_vals, buf0, buf1);
    lgcn_gather_wmma_kernel<<<gathBlk, 128, 0, stream>>>(buf1, batch_user, batch_pos,
                                                         batch_neg, out, 0);
    lgcn_zero_kernel<<<vecBlk, 256, 0, stream>>>((float4*)buf0);
    // layer 2: buf1 -> buf0
    lgcn_scatter_kernel<<<scatBlk, 256, 0, stream>>>(row_idx, col_idx, a_vals, buf1, buf0);
    lgcn_gather_wmma_kernel<<<gathBlk, 128, 0, stream>>>(buf0, batch_user, batch_pos,
                                                         batch_neg, out, 0);
    lgcn_zero_kernel<<<vecBlk, 256, 0, stream>>>((float4*)buf1);
    // layer 3: buf0 -> buf1
    lgcn_scatter_kernel<<<scatBlk, 256, 0, stream>>>(row_idx, col_idx, a_vals, buf0, buf1);
    lgcn_gather_wmma_kernel<<<gathBlk, 128, 0, stream>>>(buf1, batch_user, batch_pos,
                                                         batch_neg, out, 0);
}